// update_h_net_7395933684257
// MI455X (gfx1250) — compile-verified
//
#include <hip/hip_runtime.h>

// f32 WMMA vector types (wave32): A/B of 16x16x4_f32 = 2 VGPRs/lane, C/D = 8 VGPRs/lane
typedef __attribute__((ext_vector_type(2))) float v2f;
typedef __attribute__((ext_vector_type(8))) float v8f;

constexpr int D      = 512;   // input feature dim
constexpr int HID    = 256;   // hidden dim
constexpr int K      = 5;     // neighbors per node
constexpr int EPOCHS = 2;     // update epochs
constexpr int NT     = 8;     // 16-col tiles per wave -> 128 output cols per wave
constexpr int WAVES  = 8;     // waves per 256-thread block

// out[r, c] = sum_k A1[r,k]*W1[c,k] + b1[c]  (+ A2@W2^T + b2 if FUSE2) (+ addt[r,c] if ADDT)
// A1/A2: [nrows, DIN] row-major; W1/W2: [HID, DIN] row-major (i.e. we compute A @ W^T).
template <int DIN, bool FUSE2, bool ADDT>
__global__ __launch_bounds__(256) void gemm_wmma(
    const float* __restrict__ A1, const float* __restrict__ W1, const float* __restrict__ b1,
    const float* __restrict__ A2, const float* __restrict__ W2, const float* __restrict__ b2,
    const float* __restrict__ addt, float* __restrict__ out, int nrows)
{
    const int lane    = threadIdx.x & 31;
    const int wave    = threadIdx.x >> 5;
    const int rowTile = blockIdx.x * WAVES + wave;
    if (rowTile * 16 >= nrows) return;          // wave-uniform: EXEC stays all-ones

    const int half = lane >> 4;                 // 0: K={k,k+1}, 1: K={k+2,k+3}
    const int lr   = lane & 15;                 // M index (A) / N index (B)
    const int colBase = blockIdx.y * (16 * NT);

    const size_t arow = (size_t)(rowTile * 16 + lr) * DIN + 2 * half;
    const float* a1p = A1 + arow;
    const float* a2p = FUSE2 ? (A2 + arow) : nullptr;

    v8f acc[NT] = {};

#pragma unroll 2
    for (int kb = 0; kb < DIN; kb += 4) {
        v2f a = *(const v2f*)(a1p + kb);
#pragma unroll
        for (int t = 0; t < NT; ++t) {
            v2f b = *(const v2f*)(W1 + (size_t)(colBase + 16 * t + lr) * DIN + 2 * half + kb);
            acc[t] = __builtin_amdgcn_wmma_f32_16x16x4_f32(
                false, a, false, b, (short)0, acc[t], false, false);
        }
        if (FUSE2) {
            v2f a2 = *(const v2f*)(a2p + kb);
#pragma unroll
            for (int t = 0; t < NT; ++t) {
                v2f b2v = *(const v2f*)(W2 + (size_t)(colBase + 16 * t + lr) * DIN + 2 * half + kb);
                acc[t] = __builtin_amdgcn_wmma_f32_16x16x4_f32(
                    false, a2, false, b2v, (short)0, acc[t], false, false);
            }
        }
    }

    // C/D layout: VGPR r holds row (r + 8*half) of the 16x16 tile, col = lane%16
    const int orow0 = rowTile * 16 + 8 * half;
#pragma unroll
    for (int t = 0; t < NT; ++t) {
        const int col = colBase + 16 * t + lr;
        float bias = b1[col];
        if (FUSE2) bias += b2[col];
#pragma unroll
        for (int r = 0; r < 8; ++r) {
            const size_t oi = (size_t)(orow0 + r) * HID + col;
            float v = acc[t][r] + bias;
            if (ADDT) v += addt[oi];
            out[oi] = v;
        }
    }
}

// nm[i, :] = mean_{j<K} h[nbr[i*K+j], :]  (h fits in 192MB L2 -> gather hits L2)
__global__ __launch_bounds__(256) void gather_mean(
    const float* __restrict__ h, const int* __restrict__ nbr,
    float* __restrict__ out, int nrows)
{
    const int t    = blockIdx.x * blockDim.x + threadIdx.x;
    const int node = t >> 6;              // 64 threads per node: 256 dims / float4
    if (node >= nrows) return;
    const int d = (t & 63) * 4;

    float sx = 0.f, sy = 0.f, sz = 0.f, sw = 0.f;
#pragma unroll
    for (int j = 0; j < K; ++j) {
        const int nb = nbr[node * K + j];
        const float4 v = *(const float4*)(h + (size_t)nb * HID + d);
        sx += v.x; sy += v.y; sz += v.z; sw += v.w;
    }
    const float inv = 1.0f / (float)K;
    float4 o; o.x = sx * inv; o.y = sy * inv; o.z = sz * inv; o.w = sw * inv;
    *(float4*)(out + (size_t)node * HID + d) = o;
}

// out = (ACCUM ? out : 0) + 0.5 * relu(h)
template <bool ACCUM>
__global__ __launch_bounds__(256) void relu_comb(
    const float4* __restrict__ h, float4* __restrict__ out, int n4)
{
    const int t = blockIdx.x * blockDim.x + threadIdx.x;
    if (t >= n4) return;
    const float4 v = h[t];
    float4 o;
    o.x = fmaxf(v.x, 0.f) * 0.5f;
    o.y = fmaxf(v.y, 0.f) * 0.5f;
    o.z = fmaxf(v.z, 0.f) * 0.5f;
    o.w = fmaxf(v.w, 0.f) * 0.5f;
    if (ACCUM) {
        const float4 p = out[t];
        o.x += p.x; o.y += p.y; o.z += p.z; o.w += p.w;
    }
    out[t] = o;
}

extern "C" void kernel_launch(void* const* d_in, const int* in_sizes, int n_in,
                              void* d_out, int out_size, void* d_ws, size_t ws_size,
                              hipStream_t stream)
{
    const float* x0     = (const float*)d_in[0];
    const float* x1     = (const float*)d_in[1];
    const float* Hin0_w = (const float*)d_in[2];
    const float* Hin0_b = (const float*)d_in[3];
    const float* Hin1_w = (const float*)d_in[4];
    const float* Hin1_b = (const float*)d_in[5];
    const float* w0_w   = (const float*)d_in[6];
    const float* w0_b   = (const float*)d_in[7];
    const float* w1_w   = (const float*)d_in[8];
    const float* w1_b   = (const float*)d_in[9];
    const float* w2_w   = (const float*)d_in[10];
    const float* w2_b   = (const float*)d_in[11];
    const int*   edge0  = (const int*)d_in[12];
    const int*   edge1  = (const int*)d_in[13];
    // d_in[14]=k (=5), d_in[15]=update_epoch (=2): compile-time constants here.

    const int N = in_sizes[0] / D;                 // 100000
    const int* nbr0 = edge0 + (size_t)N * K;       // edge0[1] row
    const int* nbr1 = edge1 + (size_t)N * K;
    float* out = (float*)d_out;

    // workspace: 6 x [N, HID] f32 buffers (~615 MB total)
    const size_t bsz = (size_t)N * HID;
    float* P0 = (float*)d_ws;       // hinit0, later recycled as view-0 pong
    float* P1 = P0 + bsz;           // hinit1, later recycled as view-1 pong
    float* P2 = P1 + bsz;           // slf0 = w2(hinit1)
    float* P3 = P2 + bsz;           // slf1 = w2(hinit0)
    float* P4 = P3 + bsz;           // neighbor mean
    float* P5 = P4 + bsz;           // ping buffer

    const dim3 blk(256);
    const int rowTiles = (N + 15) / 16;
    const dim3 gGemm((rowTiles + WAVES - 1) / WAVES, HID / (16 * NT));
    const int n4 = (int)(bsz / 4);
    const dim3 gGather(((size_t)N * 64 + 255) / 256);
    const dim3 gElem((n4 + 255) / 256);

    // --- input projections: hinit_v = x_v @ Hin_v^T + b ---
    gemm_wmma<D, false, false><<<gGemm, blk, 0, stream>>>(
        x0, Hin0_w, Hin0_b, nullptr, nullptr, nullptr, nullptr, P0, N);
    gemm_wmma<D, false, false><<<gGemm, blk, 0, stream>>>(
        x1, Hin1_w, Hin1_b, nullptr, nullptr, nullptr, nullptr, P1, N);

    // --- cross-view self terms: slf_v = w2(hinit_{1-v}) ---
    gemm_wmma<HID, false, false><<<gGemm, blk, 0, stream>>>(
        P1, w2_w, w2_b, nullptr, nullptr, nullptr, nullptr, P2, N);
    gemm_wmma<HID, false, false><<<gGemm, blk, 0, stream>>>(
        P0, w2_w, w2_b, nullptr, nullptr, nullptr, nullptr, P3, N);

    // --- per-view update loops (EPOCHS=2), fused h_new = w0(h) + w1(nm) + slf ---
    {   // view 0: hinit=P0, slf=P2, pong through P5/P0
        float* src = P0; float* dst = P5;
        for (int e = 0; e < EPOCHS; ++e) {
            gather_mean<<<gGather, blk, 0, stream>>>(src, nbr0, P4, N);
            gemm_wmma<HID, true, true><<<gGemm, blk, 0, stream>>>(
                src, w0_w, w0_b, P4, w1_w, w1_b, P2, dst, N);
            float* tmp = src; src = dst; dst = tmp;   // P0 reuse OK: slf1 already built
        }
        relu_comb<false><<<gElem, blk, 0, stream>>>((const float4*)src, (float4*)out, n4);
    }
    {   // view 1: hinit=P1, slf=P3, pong through P5/P1
        float* src = P1; float* dst = P5;
        for (int e = 0; e < EPOCHS; ++e) {
            gather_mean<<<gGather, blk, 0, stream>>>(src, nbr1, P4, N);
            gemm_wmma<HID, true, true><<<gGemm, blk, 0, stream>>>(
                src, w0_w, w0_b, P4, w1_w, w1_b, P3, dst, N);
            float* tmp = src; src = dst; dst = tmp;
        }
        relu_comb<true><<<gElem, blk, 0, stream>>>((const float4*)src, (float4*)out, n4);
    }
    (void)n_in; (void)out_size; (void)ws_size;
}